// IADSAGE_7232724927268
// MI455X (gfx1250) — compile-verified
//
#include <hip/hip_runtime.h>
#include <hip/hip_bf16.h>

typedef __attribute__((ext_vector_type(2))) float v2f;
typedef __attribute__((ext_vector_type(8))) float v8f;

// ---------------------------------------------------------------------------
// Degree count: indegree per destination node (atomic on u32).
// ---------------------------------------------------------------------------
__global__ __launch_bounds__(256) void degree_count_kernel(
    const long long* __restrict__ dst, unsigned* __restrict__ cnt, int E) {
  int e = blockIdx.x * blockDim.x + threadIdx.x;
  if (e < E) atomicAdd(&cnt[(int)dst[e]], 1u);
}

// ---------------------------------------------------------------------------
// Scatter-sum of feature rows: one wave per edge, lane-strided features so
// every atomic burst is a contiguous 128B segment (coalesced in L2).
// ---------------------------------------------------------------------------
__global__ __launch_bounds__(256) void scatter_sum_kernel(
    const float* __restrict__ xin, const long long* __restrict__ src,
    const long long* __restrict__ dst, float* __restrict__ sum, int E, int F) {
  int wave = blockIdx.x * (blockDim.x >> 5) + (threadIdx.x >> 5);
  int lane = threadIdx.x & 31;
  if (wave >= E) return;
  int s = (int)src[wave];
  int d = (int)dst[wave];
  const float* xr = xin + (size_t)s * F;
  float* sr = sum + (size_t)d * F;
  for (int f = lane; f < F; f += 32) {
    atomicAdd(&sr[f], xr[f]);
  }
}

// ---------------------------------------------------------------------------
// sum -> mean (divide by clamped indegree), in place.
// ---------------------------------------------------------------------------
__global__ __launch_bounds__(256) void mean_div_kernel(
    float* __restrict__ sum, const unsigned* __restrict__ cnt, int total, int F) {
  int i = blockIdx.x * blockDim.x + threadIdx.x;
  if (i >= total) return;
  unsigned c = cnt[i / F];
  sum[i] *= (1.0f / (float)(c ? c : 1u));
}

// ---------------------------------------------------------------------------
// Fused SAGE GEMM:  out = [relu]( A1 @ W1^T + A2 @ W2^T + bias )
// A1,A2: [nrows,K] row-major; W1,W2: [O,K] row-major; out: [nrows,O].
// One wave computes a 16-row M-tile; K accumulated in steps of 4 via
// V_WMMA_F32_16X16X4_F32 (full fp32 precision, matrix pipe).
// ---------------------------------------------------------------------------
__global__ __launch_bounds__(256) void sage_gemm_wmma_kernel(
    const float* __restrict__ A1, const float* __restrict__ A2,
    const float* __restrict__ W1, const float* __restrict__ W2,
    const float* __restrict__ bias, float* __restrict__ out,
    int nrows, int K, int O, int relu) {
  const int lane = threadIdx.x & 31;
  const int wave = blockIdx.x * (blockDim.x >> 5) + (threadIdx.x >> 5);
  const int m0 = wave * 16;
  if (m0 >= nrows) return;               // wave-uniform: EXEC stays all-1s

  const int half = lane & 15;            // M (for A) / N (for B,C) sub-index
  const int hi   = lane >> 4;            // 0: K pair {k0,k0+1}; 1: {k0+2,k0+3}
  const float* a1row = A1 + (size_t)(m0 + half) * K;
  const float* a2row = A2 + (size_t)(m0 + half) * K;

  for (int n0 = 0; n0 < O; n0 += 16) {
    const float* w1n = W1 + (size_t)(n0 + half) * K;   // weight row = out-feature
    const float* w2n = W2 + (size_t)(n0 + half) * K;
    v8f acc = {};
    for (int k0 = 0; k0 < K; k0 += 4) {
      const int ka = k0 + hi * 2;
      v2f a = *(const v2f*)(a1row + ka);
      v2f b = *(const v2f*)(w1n + ka);
      acc = __builtin_amdgcn_wmma_f32_16x16x4_f32(
          false, a, false, b, (short)0, acc, false, false);
    }
    for (int k0 = 0; k0 < K; k0 += 4) {
      const int ka = k0 + hi * 2;
      v2f a = *(const v2f*)(a2row + ka);
      v2f b = *(const v2f*)(w2n + ka);
      acc = __builtin_amdgcn_wmma_f32_16x16x4_f32(
          false, a, false, b, (short)0, acc, false, false);
    }
    const float bv = bias[n0 + half];
#pragma unroll
    for (int r = 0; r < 8; ++r) {
      float v = acc[r] + bv;
      if (relu) v = fmaxf(v, 0.0f);
      const int row = m0 + r + hi * 8;   // C layout: lanes16-31 hold M = r+8
      out[(size_t)row * O + (n0 + half)] = v;
    }
  }
}

// ---------------------------------------------------------------------------
// dis[n] = rsqrt(indegree + 1)   (self loop guarantees deg >= 1)
// ---------------------------------------------------------------------------
__global__ __launch_bounds__(256) void inv_sqrt_deg_kernel(
    const unsigned* __restrict__ cnt, float* __restrict__ dis, int N) {
  int n = blockIdx.x * blockDim.x + threadIdx.x;
  if (n < N) dis[n] = rsqrtf((float)(cnt[n] + 1u));
}

// ---------------------------------------------------------------------------
// iconv self-loop term (also initializes out): out[n,f] = dis[n]^2 * h2[n,f]
// ---------------------------------------------------------------------------
__global__ __launch_bounds__(256) void iconv_init_kernel(
    const float* __restrict__ h2, const float* __restrict__ dis,
    float* __restrict__ out, int total, int C) {
  int i = blockIdx.x * blockDim.x + threadIdx.x;
  if (i >= total) return;
  float d = dis[i / C];
  out[i] = d * d * h2[i];
}

// ---------------------------------------------------------------------------
// iconv edge scatter: out[dst] += dis[src]*dis[dst] * h2[src]; wave per edge.
// ---------------------------------------------------------------------------
__global__ __launch_bounds__(256) void iconv_scatter_kernel(
    const float* __restrict__ h2, const long long* __restrict__ src,
    const long long* __restrict__ dst, const float* __restrict__ dis,
    float* __restrict__ out, int E, int C) {
  int wave = blockIdx.x * (blockDim.x >> 5) + (threadIdx.x >> 5);
  int lane = threadIdx.x & 31;
  if (wave >= E) return;
  int s = (int)src[wave];
  int d = (int)dst[wave];
  float coeff = dis[s] * dis[d];
  const float* hr = h2 + (size_t)s * C;
  float* outr = out + (size_t)d * C;
  for (int f = lane; f < C; f += 32) {
    atomicAdd(&outr[f], coeff * hr[f]);
  }
}

// ---------------------------------------------------------------------------
extern "C" void kernel_launch(void* const* d_in, const int* in_sizes, int n_in,
                              void* d_out, int out_size, void* d_ws, size_t ws_size,
                              hipStream_t stream) {
  const float*     x   = (const float*)d_in[0];
  const long long* ei  = (const long long*)d_in[1];
  const float*     W1l = (const float*)d_in[2];
  const float*     b1  = (const float*)d_in[3];
  const float*     W1r = (const float*)d_in[4];
  const float*     W2l = (const float*)d_in[5];
  const float*     b2  = (const float*)d_in[6];
  const float*     W2r = (const float*)d_in[7];
  float*           out = (float*)d_out;

  const int E = in_sizes[1] / 2;
  const int H = in_sizes[3];            // 128
  const int F = in_sizes[2] / H;        // 128
  const int C = in_sizes[6];            // 64
  const int N = in_sizes[0] / F;        // 100000

  const long long* src = ei;
  const long long* dst = ei + E;

  // Workspace carve-up (all sizes 256B-aligned).
  auto align = [](size_t v) { return (v + 255) & ~(size_t)255; };
  char* ws = (char*)d_ws;
  size_t off = 0;
  unsigned* cnt = (unsigned*)(ws + off); off += align((size_t)N * sizeof(unsigned));
  float* sum = (float*)(ws + off);       off += align((size_t)N * F * sizeof(float)); // agg buffer (F==H)
  float* h   = (float*)(ws + off);       off += align((size_t)N * H * sizeof(float));
  float* h2  = (float*)(ws + off);       off += align((size_t)N * C * sizeof(float));
  float* dis = (float*)(ws + off);       off += align((size_t)N * sizeof(float));
  (void)ws_size;

  const int B = 256;
  const int edgeWaveBlocks = (E * 32 + B - 1) / B;          // 1 wave per edge
  const int gemmBlocks     = (((N + 15) / 16) + 7) / 8;     // 8 M-tiles per block

  // ---- Layer 1: mean aggregation ----
  hipMemsetAsync(cnt, 0, (size_t)N * sizeof(unsigned), stream);
  hipMemsetAsync(sum, 0, (size_t)N * F * sizeof(float), stream);
  degree_count_kernel<<<(E + B - 1) / B, B, 0, stream>>>(dst, cnt, E);
  scatter_sum_kernel<<<edgeWaveBlocks, B, 0, stream>>>(x, src, dst, sum, E, F);
  mean_div_kernel<<<((N * F) + B - 1) / B, B, 0, stream>>>(sum, cnt, N * F, F);

  // ---- Layer 1 GEMM: h = relu(agg @ W1l^T + x @ W1r^T + b1) ----
  sage_gemm_wmma_kernel<<<gemmBlocks, B, 0, stream>>>(
      sum, x, W1l, W1r, b1, h, N, F, H, /*relu=*/1);

  // ---- Layer 2: mean aggregation of h ----
  hipMemsetAsync(sum, 0, (size_t)N * H * sizeof(float), stream);
  scatter_sum_kernel<<<edgeWaveBlocks, B, 0, stream>>>(h, src, dst, sum, E, H);
  mean_div_kernel<<<((N * H) + B - 1) / B, B, 0, stream>>>(sum, cnt, N * H, H);

  // ---- Layer 2 GEMM: h2 = agg2 @ W2l^T + h @ W2r^T + b2 ----
  sage_gemm_wmma_kernel<<<gemmBlocks, B, 0, stream>>>(
      sum, h, W2l, W2r, b2, h2, N, H, C, /*relu=*/0);

  // ---- iconv: D^-1/2 (A+I) D^-1/2 @ h2 ----
  inv_sqrt_deg_kernel<<<(N + B - 1) / B, B, 0, stream>>>(cnt, dis, N);
  iconv_init_kernel<<<((N * C) + B - 1) / B, B, 0, stream>>>(h2, dis, out, N * C, C);
  iconv_scatter_kernel<<<edgeWaveBlocks, B, 0, stream>>>(h2, src, dst, dis, out, E, C);
}